// BasicIdentifier_56822417326406
// MI455X (gfx1250) — compile-verified
//
#include <hip/hip_runtime.h>
#include <math.h>

// Problem constants (match reference)
#define NUM_TAGS   9
#define SEQ_LEN    4096
#define EMBED_DIM  512
#define BATCH      16
// log(0.5 + 1e-5)
#define LOG_THRESH_C (-0.69312721f)
#define SMOOTH     0.1f

typedef __attribute__((ext_vector_type(2))) float v2f;
typedef __attribute__((ext_vector_type(8))) float v8f;

__device__ __forceinline__ float log_sigmoid_f(float x) {
    // stable: min(x,0) - log1p(exp(-|x|))
    return fminf(x, 0.0f) - log1pf(expf(-fabsf(x)));
}

// ---------------------------------------------------------------------------
// Zero the per-(b,t) logit accumulator (256 floats) in workspace.
// ---------------------------------------------------------------------------
__global__ void zero_acc_kernel(float* __restrict__ acc) {
    acc[threadIdx.x] = 0.0f;
}

// ---------------------------------------------------------------------------
// Fused masked-weighted pooling + per-tag linear via V_WMMA_F32_16X16X4_F32.
// Per batch b:  C[16x512] = A[16x4096] * E[4096x512]
//   A[t,s] = mask[b,s] * samples[b,s,t]   (rows t>=9 are zero padding)
//   E[s,d] = emb[b,s,d]
// One wave owns one 16-wide D tile; 8 waves/block, grid = (16 batches, 4).
// Inner loop is branch-free: tag index is clamped (always-valid load) and a
// per-lane {0,1} scale zeroes the padded A rows; emb is streamed non-temporal
// (read-once, 134 MB) while samples/mask (2.9 MB, reused 32x) stay cached.
// ---------------------------------------------------------------------------
__global__ __launch_bounds__(256)
void pool_gemm_kernel(const float* __restrict__ emb,
                      const int*   __restrict__ mask,
                      const float* __restrict__ samples,
                      const float* __restrict__ W,
                      float*       __restrict__ logits_acc)
{
    const int lane = threadIdx.x & 31;
    const int wave = threadIdx.x >> 5;
    const int b    = blockIdx.x;                 // 0..15
    const int tile = blockIdx.y * 8 + wave;      // 0..31
    const int dbase = tile * 16;
    const int n  = lane & 15;                    // A row (tag t) == B col (d)
    const int kb = (lane >> 4) << 1;             // K pair: 0 (lanes 0-15), 2 (16-31)

    // Branch-free A-row handling: clamp tag column, zero-scale padded rows.
    const int   tc     = (n < NUM_TAGS) ? n : 0;
    const float tscale = (n < NUM_TAGS) ? 1.0f : 0.0f;

    const float* embB  = emb     + (size_t)b * SEQ_LEN * EMBED_DIM;
    const float* sampB = samples + (size_t)b * SEQ_LEN * (NUM_TAGS + 1);
    const int*   maskB = mask    + (size_t)b * SEQ_LEN;

    // 32-bit element offsets (SGPR base + VGPR offset addressing)
    int embOff  = kb * EMBED_DIM + dbase + n;
    int sampOff = kb * (NUM_TAGS + 1) + tc;
    int maskOff = kb;

    v8f acc = {};

    #pragma unroll 4
    for (int s0 = 0; s0 < SEQ_LEN; s0 += 4) {
        // B fragment: E[s, dbase+n] / E[s+1, dbase+n] — streamed, read-once
        v2f bfrag;
        bfrag.x = __builtin_nontemporal_load(embB + embOff);
        bfrag.y = __builtin_nontemporal_load(embB + embOff + EMBED_DIM);

        // A fragment: (tscale * mask) * samples for tag row n
        const float m0 = tscale * (float)maskB[maskOff];
        const float m1 = tscale * (float)maskB[maskOff + 1];
        v2f afrag;
        afrag.x = m0 * sampB[sampOff];
        afrag.y = m1 * sampB[sampOff + (NUM_TAGS + 1)];

        // D = A x B + C  (fp32, 16x16x4)
        acc = __builtin_amdgcn_wmma_f32_16x16x4_f32(
                  /*neg_a=*/false, afrag, /*neg_b=*/false, bfrag,
                  /*c_mod=*/(short)0, acc, /*reuse_a=*/false, /*reuse_b=*/false);

        embOff  += 4 * EMBED_DIM;
        sampOff += 4 * (NUM_TAGS + 1);
        maskOff += 4;
    }

    // C/D layout: VGPR r, lane L -> row t = r + 8*(L>>4), col d = dbase + (L&15).
    // logits[b,t] += sum_d pooled[t,d] * W[t,d]  (partial over this 16-wide tile)
    const int half = lane >> 4;
    float partial[8];
    #pragma unroll
    for (int r = 0; r < 8; ++r) {
        const int t = r + 8 * half;
        float v = (t < NUM_TAGS) ? acc[r] * W[t * EMBED_DIM + dbase + n] : 0.0f;
        // butterfly reduce across the 16 lanes sharing row t
        #pragma unroll
        for (int off = 1; off < 16; off <<= 1)
            v += __shfl_xor(v, off, 32);
        partial[r] = v;
    }
    if (n == 0) {
        #pragma unroll
        for (int r = 0; r < 8; ++r) {
            const int t = r + 8 * half;
            if (t < NUM_TAGS)
                atomicAdd(&logits_acc[b * 16 + t], partial[r]);
        }
    }
}

// ---------------------------------------------------------------------------
// Finalize: bias, log_sigmoid, preds, smoothed-BCE loss mean. One block.
// d_out layout (float32): logits[144] | log_probs[144] | preds[144] | loss[1]
// ---------------------------------------------------------------------------
__global__ __launch_bounds__(256)
void finalize_kernel(const float* __restrict__ logits_acc,
                     const float* __restrict__ bias,
                     const int*   __restrict__ labels,
                     float*       __restrict__ out)
{
    __shared__ float red[256];
    const int tid = threadIdx.x;
    float lossv = 0.0f;

    if (tid < BATCH * NUM_TAGS) {
        const int b = tid / NUM_TAGS;
        const int t = tid % NUM_TAGS;

        const float logit = logits_acc[b * 16 + t] + bias[t];
        const float lp  = log_sigmoid_f(logit);
        const float lpn = log_sigmoid_f(-logit);

        out[tid]       = logit;                                  // all_logits
        out[144 + tid] = lp;                                     // log_probs
        out[288 + tid] = (lp > LOG_THRESH_C) ? 1.0f : 0.0f;      // preds

        const float lab  = (float)labels[b * (NUM_TAGS + 1) + t];
        const float soft = lab + SMOOTH - 2.0f * SMOOTH * lab;
        lossv = -(soft * lp + (1.0f - soft) * lpn);
    }

    red[tid] = lossv;
    __syncthreads();
    #pragma unroll
    for (int sft = 128; sft > 0; sft >>= 1) {
        if (tid < sft) red[tid] += red[tid + sft];
        __syncthreads();
    }
    if (tid == 0)
        out[432] = red[0] / (float)(BATCH * NUM_TAGS);           // loss
}

// ---------------------------------------------------------------------------
extern "C" void kernel_launch(void* const* d_in, const int* in_sizes, int n_in,
                              void* d_out, int out_size, void* d_ws, size_t ws_size,
                              hipStream_t stream) {
    const float* emb     = (const float*)d_in[0];   // [16,4096,512] f32
    const int*   mask    = (const int*)  d_in[1];   // [16,4096]     i32
    const float* samples = (const float*)d_in[2];   // [16,4096,10]  f32
    const int*   labels  = (const int*)  d_in[3];   // [16,10]       i32
    const float* W       = (const float*)d_in[4];   // [9,512]       f32
    const float* bias    = (const float*)d_in[5];   // [9]           f32

    float* out = (float*)d_out;
    float* acc = (float*)d_ws;   // 256 floats: padded [16 x 16] logit accumulator

    zero_acc_kernel<<<1, 256, 0, stream>>>(acc);
    pool_gemm_kernel<<<dim3(BATCH, 4), 256, 0, stream>>>(emb, mask, samples, W, acc);
    finalize_kernel<<<1, 256, 0, stream>>>(acc, bias, labels, out);
}